// AGNN_76776835383353
// MI455X (gfx1250) — compile-verified
//
#include <hip/hip_runtime.h>

#define N_NODES  100000      // == 6250 * 16, so 16-row WMMA tiles need no row guards
#define N_EDGES  3200000
#define N_ETOT   (N_EDGES + N_NODES)   // implicit self-loops appended
#define N_GRAPHS 64

typedef __attribute__((ext_vector_type(2))) float v2f;
typedef __attribute__((ext_vector_type(8))) float v8f;

static __device__ __forceinline__ float relu_f(float v) { return v > 0.f ? v : 0.f; }

// ---------------------------------------------------------------------------
// int64 edge index -> int32 src/dst
// ---------------------------------------------------------------------------
__global__ __launch_bounds__(256) void cvt_idx_kernel(const long long* __restrict__ ei,
                                                      int* __restrict__ src32,
                                                      int* __restrict__ dst32) {
    int e = blockIdx.x * blockDim.x + threadIdx.x;
    if (e < N_EDGES) {
        src32[e] = (int)ei[e];
        dst32[e] = (int)ei[(long long)N_EDGES + e];
    }
}

// ---------------------------------------------------------------------------
// Repack weights into k-pair-interleaved layout so WMMA B operands are single
// aligned b64 loads:  Wt[(k/2)*32 + 2*col + (k&1)] = W[k][col]  (zero-padded)
// ---------------------------------------------------------------------------
__global__ __launch_bounds__(256) void pack_w1_kernel(const float* __restrict__ W1,
                                                      float* __restrict__ W1t) {
    int i = blockIdx.x * blockDim.x + threadIdx.x;   // 128*16 = 2048
    if (i < 128 * 16) {
        int k = i >> 4, c = i & 15;
        W1t[(k >> 1) * 32 + 2 * c + (k & 1)] = (c < 8) ? W1[k * 8 + c] : 0.f;
    }
}

__global__ __launch_bounds__(128) void pack_w2_kernel(const float* __restrict__ W2,
                                                      float* __restrict__ W2t) {
    int i = threadIdx.x;                              // 8*16 = 128
    if (i < 8 * 16) {
        int k = i >> 4, c = i & 15;
        W2t[(k >> 1) * 32 + 2 * c + (k & 1)] = W2[k * 16 + c];
    }
}

// ---------------------------------------------------------------------------
// h0 = relu(x @ W1 + b1)   [N,128] x [128,8->16pad] -> [N,8]
// One wave32 computes a 16-row tile via V_WMMA_F32_16X16X4_F32.
// A layout: row = m0 + lane%16, K = vgpr + 2*(lane/16) within each x4 step.
// C layout: M = v + 8*(lane/16), N = lane%16.
// ---------------------------------------------------------------------------
__global__ __launch_bounds__(256) void lin1_wmma_kernel(const float* __restrict__ x,
                                                        const float* __restrict__ W1t,
                                                        const float* __restrict__ b1,
                                                        float* __restrict__ h0) {
    int wave = (blockIdx.x * blockDim.x + threadIdx.x) >> 5;
    int lane = threadIdx.x & 31;
    int m0 = wave * 16;
    if (m0 >= N_NODES) return;

    int row = m0 + (lane & 15);
    int kh  = lane >> 4;          // 0 or 1 -> K-half inside the x4 step
    int col = lane & 15;

    const float* xr = x + (size_t)row * 128 + 2 * kh;     // A: k, k+1 contiguous
    const float* wc = W1t + kh * 32 + 2 * col;            // B: k, k+1 contiguous

    v8f c = {};
#pragma unroll
    for (int k0 = 0; k0 < 128; k0 += 4) {
        v2f a = *(const v2f*)(xr + k0);
        v2f b = *(const v2f*)(wc + (k0 >> 1) * 32);
        c = __builtin_amdgcn_wmma_f32_16x16x4_f32(false, a, false, b, (short)0, c,
                                                  false, false);
    }
    if (col < 8) {
        float bias = b1[col];
#pragma unroll
        for (int v = 0; v < 8; ++v) {
            int r = m0 + v + 8 * kh;
            h0[r * 8 + col] = relu_f(c[v] + bias);
        }
    }
}

// ---------------------------------------------------------------------------
// Per-node: hn = h / max(||h||, eps); init nsum=0, hout=0
// ---------------------------------------------------------------------------
__global__ __launch_bounds__(256) void norm_kernel(const float* __restrict__ hin,
                                                   float* __restrict__ hn,
                                                   float* __restrict__ nsum,
                                                   float* __restrict__ hout) {
    int i = blockIdx.x * blockDim.x + threadIdx.x;
    if (i >= N_NODES) return;
    const float4 lo = *(const float4*)(hin + i * 8);
    const float4 hi = *(const float4*)(hin + i * 8 + 4);
    float ss = lo.x * lo.x + lo.y * lo.y + lo.z * lo.z + lo.w * lo.w +
               hi.x * hi.x + hi.y * hi.y + hi.z * hi.z + hi.w * hi.w;
    float inv = 1.f / fmaxf(sqrtf(ss), 1e-12f);
    float4 nlo = make_float4(lo.x * inv, lo.y * inv, lo.z * inv, lo.w * inv);
    float4 nhi = make_float4(hi.x * inv, hi.y * inv, hi.z * inv, hi.w * inv);
    *(float4*)(hn + i * 8)     = nlo;
    *(float4*)(hn + i * 8 + 4) = nhi;
    nsum[i] = 0.f;
    float4 z = make_float4(0.f, 0.f, 0.f, 0.f);
    *(float4*)(hout + i * 8)     = z;
    *(float4*)(hout + i * 8 + 4) = z;
}

// ---------------------------------------------------------------------------
// Single fused edge pass (softmax needs no max-shift: |alpha| <= |beta|):
//   a = exp(beta * dot(hn[d], hn[s]));  nsum[d] += a;  hout[d] += a * hin[s]
// Normalization by nsum deferred to div_kernel (sum(a*h)/s == sum((a/s)*h)).
// ---------------------------------------------------------------------------
__global__ __launch_bounds__(256) void edge_kernel(const float* __restrict__ hn,
                                                   const float* __restrict__ hin,
                                                   const int* __restrict__ src32,
                                                   const int* __restrict__ dst32,
                                                   const float* __restrict__ beta_ptr,
                                                   float* __restrict__ nsum,
                                                   float* __restrict__ hout) {
    int e = blockIdx.x * blockDim.x + threadIdx.x;
    if (e >= N_ETOT) return;
    int s, d;
    if (e < N_EDGES) { s = src32[e]; d = dst32[e]; }
    else             { s = d = e - N_EDGES; }
    const float4 a0 = *(const float4*)(hn + s * 8);
    const float4 a1 = *(const float4*)(hn + s * 8 + 4);
    const float4 b0 = *(const float4*)(hn + d * 8);
    const float4 b1 = *(const float4*)(hn + d * 8 + 4);
    float dot = a0.x * b0.x + a0.y * b0.y + a0.z * b0.z + a0.w * b0.w +
                a1.x * b1.x + a1.y * b1.y + a1.z * b1.z + a1.w * b1.w;
    float beta = beta_ptr ? beta_ptr[0] : 1.0f;
    float a = expf(beta * dot);
    atomicAdd(nsum + d, a);
    const float4 v0 = *(const float4*)(hin + s * 8);
    const float4 v1 = *(const float4*)(hin + s * 8 + 4);
    float* o = hout + d * 8;
    atomicAdd(o + 0, v0.x * a); atomicAdd(o + 1, v0.y * a);
    atomicAdd(o + 2, v0.z * a); atomicAdd(o + 3, v0.w * a);
    atomicAdd(o + 4, v1.x * a); atomicAdd(o + 5, v1.y * a);
    atomicAdd(o + 6, v1.z * a); atomicAdd(o + 7, v1.w * a);
}

// ---------------------------------------------------------------------------
// hout[i] /= nsum[i]   (nsum > 0 guaranteed: self-loop contributes exp(..)>0)
// ---------------------------------------------------------------------------
__global__ __launch_bounds__(256) void div_kernel(float* __restrict__ hout,
                                                  const float* __restrict__ nsum) {
    int i = blockIdx.x * blockDim.x + threadIdx.x;
    if (i >= N_NODES) return;
    float inv = 1.f / nsum[i];
    float4 lo = *(const float4*)(hout + i * 8);
    float4 hi = *(const float4*)(hout + i * 8 + 4);
    lo.x *= inv; lo.y *= inv; lo.z *= inv; lo.w *= inv;
    hi.x *= inv; hi.y *= inv; hi.z *= inv; hi.w *= inv;
    *(float4*)(hout + i * 8)     = lo;
    *(float4*)(hout + i * 8 + 4) = hi;
}

// ---------------------------------------------------------------------------
// hrel = relu(h @ W2 + b2)   [N,8] x [8,16] -> [N,16]  (K=8 -> two x4 steps)
// ---------------------------------------------------------------------------
__global__ __launch_bounds__(256) void lin2_wmma_kernel(const float* __restrict__ h,
                                                        const float* __restrict__ W2t,
                                                        const float* __restrict__ b2,
                                                        float* __restrict__ hrel) {
    int wave = (blockIdx.x * blockDim.x + threadIdx.x) >> 5;
    int lane = threadIdx.x & 31;
    int m0 = wave * 16;
    if (m0 >= N_NODES) return;

    int row = m0 + (lane & 15);
    int kh  = lane >> 4;
    int col = lane & 15;

    const float* hr = h + (size_t)row * 8 + 2 * kh;
    const float* wc = W2t + kh * 32 + 2 * col;

    v8f c = {};
#pragma unroll
    for (int k0 = 0; k0 < 8; k0 += 4) {
        v2f a = *(const v2f*)(hr + k0);
        v2f b = *(const v2f*)(wc + (k0 >> 1) * 32);
        c = __builtin_amdgcn_wmma_f32_16x16x4_f32(false, a, false, b, (short)0, c,
                                                  false, false);
    }
    float bias = b2[col];
#pragma unroll
    for (int v = 0; v < 8; ++v) {
        int r = m0 + v + 8 * kh;
        hrel[r * 16 + col] = relu_f(c[v] + bias);
    }
}

// ---------------------------------------------------------------------------
// Pooling init + per-node pooling atomics + final tiny GEMM
// ---------------------------------------------------------------------------
__global__ __launch_bounds__(1024) void pool_init_kernel(float* __restrict__ gmax,
                                                         float* __restrict__ gsum,
                                                         float* __restrict__ cnt) {
    int i = threadIdx.x;
    if (i < N_GRAPHS * 16) { gmax[i] = -__builtin_inff(); gsum[i] = 0.f; }
    if (i < N_GRAPHS) cnt[i] = 0.f;
}

__global__ __launch_bounds__(256) void pool_kernel(const float* __restrict__ hrel,
                                                   const long long* __restrict__ batch,
                                                   float* __restrict__ gmax,
                                                   float* __restrict__ gsum,
                                                   float* __restrict__ cnt) {
    int i = blockIdx.x * blockDim.x + threadIdx.x;
    if (i >= N_NODES) return;
    int g = (int)batch[i];
    atomicAdd(cnt + g, 1.0f);
#pragma unroll
    for (int k = 0; k < 16; ++k) {
        float v = hrel[i * 16 + k];
        atomicMax(gmax + g * 16 + k, v);
        atomicAdd(gsum + g * 16 + k, v);
    }
}

__global__ __launch_bounds__(128) void final_kernel(const float* __restrict__ gmax,
                                                    const float* __restrict__ gsum,
                                                    const float* __restrict__ cnt,
                                                    const float* __restrict__ W3,
                                                    const float* __restrict__ b3,
                                                    float* __restrict__ out) {
    int t = threadIdx.x;
    if (t >= N_GRAPHS * 2) return;
    int g = t >> 1, c = t & 1;
    float cn = fmaxf(cnt[g], 1.0f);
    float acc = b3[c];
#pragma unroll
    for (int j = 0; j < 16; ++j) {
        float gm = gmax[g * 16 + j];
        if (!(gm > -__builtin_inff())) gm = 0.f;   // isfinite-or-zero (catch -inf/NaN)
        acc += gm * W3[j * 2 + c];
    }
#pragma unroll
    for (int j = 0; j < 16; ++j)
        acc += (gsum[g * 16 + j] / cn) * W3[(16 + j) * 2 + c];
    out[g * 2 + c] = acc;
}

// ---------------------------------------------------------------------------
extern "C" void kernel_launch(void* const* d_in, const int* in_sizes, int n_in,
                              void* d_out, int out_size, void* d_ws, size_t ws_size,
                              hipStream_t stream) {
    const float*     x     = (const float*)d_in[0];
    const long long* ei    = (const long long*)d_in[1];
    const long long* batch = (const long long*)d_in[2];
    const float*     W1    = (const float*)d_in[3];
    const float*     b1    = (const float*)d_in[4];
    const float*     beta2 = (const float*)d_in[5];
    const float*     W2    = (const float*)d_in[6];
    const float*     b2    = (const float*)d_in[7];
    const float*     W3    = (const float*)d_in[8];
    const float*     b3    = (const float*)d_in[9];
    float* out = (float*)d_out;

    char* ws = (char*)d_ws;
    size_t off = 0;
    auto carve = [&](size_t bytes) {
        char* p = ws + off;
        off += (bytes + 255) & ~(size_t)255;
        return p;
    };
    int*   src32 = (int*)  carve((size_t)N_EDGES * 4);
    int*   dst32 = (int*)  carve((size_t)N_EDGES * 4);
    float* h0    = (float*)carve((size_t)N_NODES * 8 * 4);   // lin1 out / conv2 out
    float* hA    = (float*)carve((size_t)N_NODES * 8 * 4);   // conv1 out
    float* hn    = (float*)carve((size_t)N_NODES * 8 * 4);   // normalized features
    float* nsum  = (float*)carve((size_t)N_NODES * 4);
    float* hrel  = (float*)carve((size_t)N_NODES * 16 * 4);  // lin2 out
    float* gmax  = (float*)carve((size_t)N_GRAPHS * 16 * 4);
    float* gsum  = (float*)carve((size_t)N_GRAPHS * 16 * 4);
    float* cnt   = (float*)carve((size_t)N_GRAPHS * 4);
    float* W1t   = (float*)carve((size_t)128 * 16 * 4);      // packed/padded W1
    float* W2t   = (float*)carve((size_t)8 * 16 * 4);        // packed W2

    const int B = 256;
    int gridE32 = (N_EDGES + B - 1) / B;
    int gridET  = (N_ETOT + B - 1) / B;
    int gridN   = (N_NODES + B - 1) / B;
    int nWaves  = N_NODES / 16;                 // exact (100000 % 16 == 0)
    int gridW   = (nWaves + (B / 32) - 1) / (B / 32);

    cvt_idx_kernel<<<gridE32, B, 0, stream>>>(ei, src32, dst32);
    pack_w1_kernel<<<(128 * 16 + B - 1) / B, B, 0, stream>>>(W1, W1t);
    pack_w2_kernel<<<1, 128, 0, stream>>>(W2, W2t);
    lin1_wmma_kernel<<<gridW, B, 0, stream>>>(x, W1t, b1, h0);

    // conv1: h0 -> hA  (beta fixed = 1.0)
    norm_kernel<<<gridN,  B, 0, stream>>>(h0, hn, nsum, hA);
    edge_kernel<<<gridET, B, 0, stream>>>(hn, h0, src32, dst32,
                                          (const float*)nullptr, nsum, hA);
    div_kernel <<<gridN,  B, 0, stream>>>(hA, nsum);

    // conv2: hA -> h0  (learnable beta2)
    norm_kernel<<<gridN,  B, 0, stream>>>(hA, hn, nsum, h0);
    edge_kernel<<<gridET, B, 0, stream>>>(hn, hA, src32, dst32, beta2, nsum, h0);
    div_kernel <<<gridN,  B, 0, stream>>>(h0, nsum);

    lin2_wmma_kernel<<<gridW, B, 0, stream>>>(h0, W2t, b2, hrel);

    pool_init_kernel<<<1, 1024, 0, stream>>>(gmax, gsum, cnt);
    pool_kernel     <<<gridN, B, 0, stream>>>(hrel, batch, gmax, gsum, cnt);
    final_kernel    <<<1, 128, 0, stream>>>(gmax, gsum, cnt, W3, b3, out);
}